// CRAP_16544214024675
// MI455X (gfx1250) — compile-verified
//
#include <hip/hip_runtime.h>
#include <hip/hip_bf16.h>
#include <math.h>

typedef __attribute__((ext_vector_type(2))) float v2f;
typedef __attribute__((ext_vector_type(8))) float v8f;

#define BATCH 4
#define CCH   256
#define D1    4096    // 64*64
#define D2    16384   // 128*128

// Y[b] = W(256x256) @ X[b](256xd) + bias, fp32 via V_WMMA_F32_16X16X4_F32.
// Block = 128 threads (4 waves). Each wave owns one 16x64 output strip
// (4 N-tiles), sharing A fragments across the 4 WMMAs per K-step.
// grid.x = B * 16 * (d/256)
__global__ void __launch_bounds__(128)
gemm256_wmma_f32(const float* __restrict__ W,
                 const float* __restrict__ X,
                 const float* __restrict__ bias,
                 float* __restrict__ Y,
                 int d)
{
    const int ngrps = d >> 8;                 // groups of 256 columns
    int flat  = blockIdx.x;
    int ngrp  = flat % ngrps;
    int rest  = flat / ngrps;
    int mtile = rest & 15;                    // 16 row-tiles of 16
    int b     = rest >> 4;

    int lane = threadIdx.x & 31;
    int wave = threadIdx.x >> 5;              // 0..3
    int lmod = lane & 15;
    int lhi  = lane >> 4;                     // 0 or 1

    int m0 = mtile * 16;
    int n0 = ngrp * 256 + wave * 64;          // this wave's 64-col strip

    // A: lane holds W[m0+lmod, k0 + 2*lhi + {0,1}]  (contiguous 8B)
    const float* Wp = W + (size_t)(m0 + lmod) * CCH + 2 * lhi;
    // B: lane holds X[k0 + 2*lhi + {0,1}, n] (two rows, stride d)
    const float* Xp = X + (size_t)b * CCH * d + (size_t)(2 * lhi) * d + (n0 + lmod);

    v8f acc0 = {}, acc1 = {}, acc2 = {}, acc3 = {};

    #pragma unroll 4
    for (int k0 = 0; k0 < CCH; k0 += 4) {
        v2f a = *(const v2f*)(Wp + k0);       // 8B-aligned (k0%4==0, 2*lhi even)
        const float* xr0 = Xp + (size_t)k0 * d;
        const float* xr1 = xr0 + d;
        v2f b0, b1, b2, b3;
        b0.x = xr0[0];  b0.y = xr1[0];
        b1.x = xr0[16]; b1.y = xr1[16];
        b2.x = xr0[32]; b2.y = xr1[32];
        b3.x = xr0[48]; b3.y = xr1[48];
        acc0 = __builtin_amdgcn_wmma_f32_16x16x4_f32(false, a, false, b0, (short)0, acc0, false, false);
        acc1 = __builtin_amdgcn_wmma_f32_16x16x4_f32(false, a, false, b1, (short)0, acc1, false, false);
        acc2 = __builtin_amdgcn_wmma_f32_16x16x4_f32(false, a, false, b2, (short)0, acc2, false, false);
        acc3 = __builtin_amdgcn_wmma_f32_16x16x4_f32(false, a, false, b3, (short)0, acc3, false, false);
    }

    // D: lane L, elem v -> row m0 + v + 8*lhi, col n + lmod
    float* Yp = Y + (size_t)b * CCH * d;
    #pragma unroll
    for (int v = 0; v < 8; ++v) {
        int m = m0 + v + 8 * lhi;
        float bv = bias[m];
        size_t row = (size_t)m * d;
        Yp[row + n0 +  0 + lmod] = acc0[v] + bv;
        Yp[row + n0 + 16 + lmod] = acc1[v] + bv;
        Yp[row + n0 + 32 + lmod] = acc2[v] + bv;
        Yp[row + n0 + 48 + lmod] = acc3[v] + bv;
    }
}

// Per (b,c): logits[k=(i,j)] = sum_{h,w} q[h,w] * src[2h+i-1, 2w+j-1]
// (zero outside), then softmax over the 9 positions with scale 1/64.
// One block (256 threads) per (b,c).
__global__ void __launch_bounds__(256)
attn_softmax(const float* __restrict__ q,     // (B*C, 4096)
             const float* __restrict__ src,   // (B*C, 128*128)
             float* __restrict__ A)           // (B*C, 9)
{
    int bc = blockIdx.x;
    const float* qp = q   + (size_t)bc * D1;
    const float* sp = src + (size_t)bc * D2;

    float part[9];
    #pragma unroll
    for (int k = 0; k < 9; ++k) part[k] = 0.f;

    for (int hw = threadIdx.x; hw < D1; hw += 256) {
        int h = hw >> 6, w = hw & 63;
        float qv = qp[hw];
        int r0 = 2 * h - 1, c0 = 2 * w - 1;   // max index 127, only low OOB possible
        #pragma unroll
        for (int i = 0; i < 3; ++i) {
            int r = r0 + i;
            #pragma unroll
            for (int j = 0; j < 3; ++j) {
                int c = c0 + j;
                float s = (r >= 0 && c >= 0) ? sp[r * 128 + c] : 0.f;
                part[3 * i + j] += qv * s;
            }
        }
    }

    __shared__ float red[9][256];
    #pragma unroll
    for (int k = 0; k < 9; ++k) red[k][threadIdx.x] = part[k];
    __syncthreads();
    for (int s = 128; s > 0; s >>= 1) {
        if (threadIdx.x < (unsigned)s) {
            #pragma unroll
            for (int k = 0; k < 9; ++k)
                red[k][threadIdx.x] += red[k][threadIdx.x + s];
        }
        __syncthreads();
    }

    if (threadIdx.x == 0) {
        const float scale = 1.0f / 64.0f;     // 1/sqrt(4096)
        float l[9], mx = -INFINITY;
        #pragma unroll
        for (int k = 0; k < 9; ++k) { l[k] = red[k][0] * scale; mx = fmaxf(mx, l[k]); }
        float sum = 0.f;
        #pragma unroll
        for (int k = 0; k < 9; ++k) { l[k] = expf(l[k] - mx); sum += l[k]; }
        float inv = 1.0f / sum;
        #pragma unroll
        for (int k = 0; k < 9; ++k) A[(size_t)bc * 9 + k] = l[k] * inv;
    }
}

// out[y,x] = feat[y,x] * sum_{i,j} [0<=y+1-i<64][0<=x+1-j<64]
//                        * A[3i+j] * vfull[2y+1-i, 2x+1-j]
__global__ void __launch_bounds__(256)
fold_combine(const float* __restrict__ feat,   // (B*C, 4096)
             const float* __restrict__ vfull,  // (B*C, 128*128)
             const float* __restrict__ A,      // (B*C, 9)
             float* __restrict__ out)          // (B*C, 4096)
{
    int idx = blockIdx.x * 256 + threadIdx.x;  // block stays within one (b,c)
    int bc = idx >> 12;
    int hw = idx & 4095;
    int y = hw >> 6, x = hw & 63;

    const float* vp = vfull + (size_t)bc * D2;
    const float* Ap = A + (size_t)bc * 9;

    float acc = 0.f;
    #pragma unroll
    for (int i = 0; i < 3; ++i) {
        int hy = y + 1 - i;
        if (hy < 0 || hy > 63) continue;       // fold validity (NOT just v-index)
        int r = 2 * y + 1 - i;
        #pragma unroll
        for (int j = 0; j < 3; ++j) {
            int wx = x + 1 - j;
            if (wx < 0 || wx > 63) continue;
            int c = 2 * x + 1 - j;
            acc += Ap[3 * i + j] * vp[r * 128 + c];
        }
    }
    out[idx] = acc * feat[idx];
}

extern "C" void kernel_launch(void* const* d_in, const int* in_sizes, int n_in,
                              void* d_out, int out_size, void* d_ws, size_t ws_size,
                              hipStream_t stream) {
    (void)in_sizes; (void)n_in; (void)out_size; (void)ws_size;
    const float* feat = (const float*)d_in[0];  // (4,256,64,64)
    const float* src  = (const float*)d_in[1];  // (4,256,128,128)
    const float* Wq   = (const float*)d_in[2];  // (256,256)
    const float* bq   = (const float*)d_in[3];  // (256,)
    const float* Wv   = (const float*)d_in[4];  // (256,256)
    const float* bv   = (const float*)d_in[5];  // (256,)
    float* out = (float*)d_out;

    float* qws = (float*)d_ws;                          // 4*256*4096  = 16 MB
    float* vws = qws + (size_t)BATCH * CCH * D1;        // 4*256*16384 = 64 MB
    float* Aws = vws + (size_t)BATCH * CCH * D2;        // 4*256*9

    // q = Wq @ feat + bq          (d = 4096)
    gemm256_wmma_f32<<<BATCH * 16 * (D1 / 256), 128, 0, stream>>>(Wq, feat, bq, qws, D1);
    // vfull = Wv @ src + bv       (d = 16384)
    gemm256_wmma_f32<<<BATCH * 16 * (D2 / 256), 128, 0, stream>>>(Wv, src, bv, vws, D2);
    // attention logits (vs raw src) + softmax over 9 positions
    attn_softmax<<<BATCH * CCH, 256, 0, stream>>>(qws, src, Aws);
    // fused fold(A*unfold(vfull)) * feat
    fold_combine<<<BATCH * CCH * D1 / 256, 256, 0, stream>>>(feat, vws, Aws, out);
}